// GraphNN_87110526697700
// MI455X (gfx1250) — compile-verified
//
#include <hip/hip_runtime.h>
#include <hip/hip_bf16.h>

#define D 128
#define DPAD 132   // LDS row stride in floats: bank-conflict-free A reads,
                   // and 132*4B per row keeps float4 (16B) alignment.

typedef __attribute__((ext_vector_type(2))) float v2f;
typedef __attribute__((ext_vector_type(8))) float v8f;

// ---------------------------------------------------------------------------
// Zero the segment-sum accumulator (N*D floats) and the counter (N floats),
// laid out contiguously so one grid-stride pass covers both.
// ---------------------------------------------------------------------------
__global__ void zero_kernel(float* __restrict__ p, long long n) {
    long long i = (long long)blockIdx.x * blockDim.x + threadIdx.x;
    long long stride = (long long)gridDim.x * blockDim.x;
    for (; i < n; i += stride) p[i] = 0.0f;
}

// ---------------------------------------------------------------------------
// Node transform: s = x @ Ws + bs ; t = x @ Wd + bd   (full fp32 via WMMA)
// Block = 256 threads = 8 waves; one 16-node row strip per block.
// Each wave owns one 16-wide column tile (8 tiles cover D=128) and runs the
// K loop in steps of 4 with V_WMMA_F32_16X16X4_F32 for both weight matrices.
// ---------------------------------------------------------------------------
__global__ void __launch_bounds__(256)
xform_kernel(const float* __restrict__ x,
             const float* __restrict__ Ws, const float* __restrict__ bs,
             const float* __restrict__ Wd, const float* __restrict__ bd,
             float* __restrict__ s_out, float* __restrict__ t_out, int N) {
    __shared__ float xt[16 * DPAD];

    const int row0 = blockIdx.x * 16;
    const int tid  = threadIdx.x;
    const bool fullTile = (row0 + 16 <= N);

    // Cooperative load of the 16 x 128 x-tile into LDS (float4 granularity):
    // 512 float4s, 256 threads -> 2 each.
    for (int i = tid; i < 16 * (D / 4); i += 256) {
        const int r = i >> 5;              // 0..15
        const int c4 = (i & 31) * 4;       // 0,4,...,124
        const int gr = row0 + r;
        float4 v;
        if (fullTile || gr < N) {
            v = *(const float4*)(x + (long long)gr * D + c4);
        } else {
            v = make_float4(0.f, 0.f, 0.f, 0.f);
        }
        *(float4*)(&xt[r * DPAD + c4]) = v;
    }
    __syncthreads();

    const int wave = tid >> 5;         // 0..7 -> column tile
    const int lane = tid & 31;
    const int ncol = wave * 16;

    // A fragment (16x4 f32): lane&15 = M row; (lane>>4)*2 = K base (V0=K, V1=K+1)
    const int m  = lane & 15;
    const int kb = (lane >> 4) * 2;
    // B fragment (4x16 f32): lane&15 = N col; (lane>>4)*2 = K base (V0=K, V1=K+1)
    const int nc = ncol + (lane & 15);

    v8f accS = {};
    v8f accT = {};

    const float* __restrict__ xr = &xt[m * DPAD];
    for (int k = 0; k < D; k += 4) {
        v2f a;
        a.x = xr[k + kb];
        a.y = xr[k + kb + 1];
        v2f bS;
        bS.x = Ws[(long long)(k + kb) * D + nc];
        bS.y = Ws[(long long)(k + kb + 1) * D + nc];
        v2f bT;
        bT.x = Wd[(long long)(k + kb) * D + nc];
        bT.y = Wd[(long long)(k + kb + 1) * D + nc];
        accS = __builtin_amdgcn_wmma_f32_16x16x4_f32(
            false, a, false, bS, (short)0, accS, false, false);
        accT = __builtin_amdgcn_wmma_f32_16x16x4_f32(
            false, a, false, bT, (short)0, accT, false, false);
    }

    const float biasS = bs[nc];
    const float biasT = bd[nc];
    // C/D layout: VGPR r -> row (r + 8*(lane>>4)), col = lane&15 within tile.
    const int mBase = (lane >> 4) * 8;
    float* __restrict__ sp = s_out + (long long)(row0 + mBase) * D + nc;
    float* __restrict__ tp = t_out + (long long)(row0 + mBase) * D + nc;
    if (fullTile) {
        #pragma unroll
        for (int r = 0; r < 8; ++r) {
            sp[(long long)r * D] = accS[r] + biasS;
            tp[(long long)r * D] = accT[r] + biasT;
        }
    } else {
        for (int r = 0; r < 8; ++r) {
            if (row0 + mBase + r < N) {
                sp[(long long)r * D] = accS[r] + biasS;
                tp[(long long)r * D] = accT[r] + biasT;
            }
        }
    }
}

// ---------------------------------------------------------------------------
// Edge message + segment sum: one wave32 per edge, 4 floats per lane.
// msg = relu(s[src] + t[dest] + ee[class]); atomically accumulate into
// sum[dest] (L2-resident, 25.6 MB) and bump cnt[dest].
// ---------------------------------------------------------------------------
__global__ void __launch_bounds__(256)
edge_kernel(const float* __restrict__ s, const float* __restrict__ t,
            const float* __restrict__ ee,
            const int* __restrict__ src, const int* __restrict__ dst,
            const int* __restrict__ cls,
            float* __restrict__ sum, float* __restrict__ cnt, int E) {
    const int e    = (int)(((long long)blockIdx.x * blockDim.x + threadIdx.x) >> 5);
    const int lane = threadIdx.x & 31;
    if (e >= E) return;

    const int a = src[e];
    const int b = dst[e];
    const int c = cls[e];

    const float4 sv = ((const float4*)(s  + (long long)a * D))[lane];
    const float4 tv = ((const float4*)(t  + (long long)b * D))[lane];
    const float4 ev = ((const float4*)(ee + (long long)c * D))[lane];

    float4 msg;
    msg.x = fmaxf(sv.x + tv.x + ev.x, 0.0f);
    msg.y = fmaxf(sv.y + tv.y + ev.y, 0.0f);
    msg.z = fmaxf(sv.z + tv.z + ev.z, 0.0f);
    msg.w = fmaxf(sv.w + tv.w + ev.w, 0.0f);

    float* dsum = sum + (long long)b * D + lane * 4;
    atomicAdd(dsum + 0, msg.x);
    atomicAdd(dsum + 1, msg.y);
    atomicAdd(dsum + 2, msg.z);
    atomicAdd(dsum + 3, msg.w);
    if (lane == 0) atomicAdd(cnt + b, 1.0f);
}

// ---------------------------------------------------------------------------
// Finalize: mean (TF unsorted_segment_mean semantics), residual, LayerNorm.
// One wave32 per node; wave reductions via __shfl_xor (wave32-native).
// Alias-safe for out == x (lane reads its elements into regs before writing).
// ---------------------------------------------------------------------------
__global__ void __launch_bounds__(256)
finalize_kernel(const float* __restrict__ x, const float* __restrict__ sum,
                const float* __restrict__ cnt,
                const float* __restrict__ gamma, const float* __restrict__ beta,
                float* __restrict__ out, int N) {
    const int node = (int)(((long long)blockIdx.x * blockDim.x + threadIdx.x) >> 5);
    const int lane = threadIdx.x & 31;
    if (node >= N) return;

    const float c   = cnt[node];
    const float inv = (c > 0.0f) ? (1.0f / c) : 0.0f;

    const float4 sv = ((const float4*)(sum + (long long)node * D))[lane];
    const float4 xv = ((const float4*)(x   + (long long)node * D))[lane];

    float y0 = sv.x * inv + xv.x;
    float y1 = sv.y * inv + xv.y;
    float y2 = sv.z * inv + xv.z;
    float y3 = sv.w * inv + xv.w;

    float ls = y0 + y1 + y2 + y3;
    for (int off = 16; off > 0; off >>= 1) ls += __shfl_xor(ls, off, 32);
    const float mu = ls * (1.0f / (float)D);

    const float d0 = y0 - mu, d1 = y1 - mu, d2 = y2 - mu, d3 = y3 - mu;
    float vs = d0 * d0 + d1 * d1 + d2 * d2 + d3 * d3;
    for (int off = 16; off > 0; off >>= 1) vs += __shfl_xor(vs, off, 32);
    const float rinv = rsqrtf(vs * (1.0f / (float)D) + 1e-3f);

    const float4 gv = ((const float4*)gamma)[lane];
    const float4 bv = ((const float4*)beta)[lane];

    float4 o;
    o.x = gv.x * d0 * rinv + bv.x;
    o.y = gv.y * d1 * rinv + bv.y;
    o.z = gv.z * d2 * rinv + bv.z;
    o.w = gv.w * d3 * rinv + bv.w;
    ((float4*)(out + (long long)node * D))[lane] = o;
}

// ---------------------------------------------------------------------------
// Host-side orchestration (graph-capture safe: launches only).
// ---------------------------------------------------------------------------
extern "C" void kernel_launch(void* const* d_in, const int* in_sizes, int n_in,
                              void* d_out, int out_size, void* d_ws, size_t ws_size,
                              hipStream_t stream) {
    const float* x      = (const float*)d_in[0];
    const int*   src    = (const int*)  d_in[1];
    const int*   dst    = (const int*)  d_in[2];
    const int*   cls    = (const int*)  d_in[3];
    const float* W_src  = (const float*)d_in[4];
    const float* b_src  = (const float*)d_in[5];
    const float* W_dest = (const float*)d_in[6];
    const float* b_dest = (const float*)d_in[7];
    const float* ee     = (const float*)d_in[8];
    const float* gam    = (const float*)d_in[9];
    const float* bet    = (const float*)d_in[10];

    const int N  = in_sizes[0] / D;
    const int E  = in_sizes[1];
    const int H  = in_sizes[4] / (D * D);
    const long long eeHop = (long long)in_sizes[8] / H;   // NE * D

    const long long nd   = (long long)N * D;
    const long long npad = ((long long)N + 63) & ~63LL;   // keep 256B alignment

    float* ws   = (float*)d_ws;
    float* sBuf = ws;                 // [N*D]
    float* tBuf = sBuf + nd;          // [N*D]
    float* sum  = tBuf + nd;          // [N*D]
    float* cntb = sum + nd;           // [N]   (contiguous after sum -> one zero pass)
    float* xtmp = cntb + npad;        // [N*D]

    const int nodeTiles = (N + 15) / 16;
    const int edgeBlks  = (E + 7) / 8;          // 8 edges (waves) per 256-thr block
    const int nodeBlks  = (N + 7) / 8;          // 8 nodes (waves) per 256-thr block
    const long long zN  = nd + N;
    int zeroBlks = (int)((zN + 255) / 256);
    if (zeroBlks > 4096) zeroBlks = 4096;

    for (int h = 0; h < H; ++h) {
        const float* xin  = (h == 0) ? x : ((h == 1) ? xtmp : (float*)d_out);
        float*       xout = (h == 0 && H > 1) ? xtmp : (float*)d_out;

        zero_kernel<<<zeroBlks, 256, 0, stream>>>(sum, zN);

        xform_kernel<<<nodeTiles, 256, 0, stream>>>(
            xin,
            W_src  + (long long)h * D * D, b_src  + (long long)h * D,
            W_dest + (long long)h * D * D, b_dest + (long long)h * D,
            sBuf, tBuf, N);

        edge_kernel<<<edgeBlks, 256, 0, stream>>>(
            sBuf, tBuf, ee + (long long)h * eeHop,
            src, dst, cls, sum, cntb, E);

        finalize_kernel<<<nodeBlks, 256, 0, stream>>>(
            xin, sum, cntb,
            gam + (long long)h * D, bet + (long long)h * D,
            xout, N);
    }
}